// CKConv_85950885527678
// MI455X (gfx1250) — compile-verified
//
#include <hip/hip_runtime.h>

#define B_    8
#define CI    32
#define CO    32
#define T_    2048
#define D_    32
#define L_    2049
#define LPAD  2052   // Kfl row stride (floats), multiple of 4 -> 16B-aligned chunk starts
#define TT    64     // time tile per block
#define DC    64     // d-chunk per LDS stage
#define KST   68     // padded LDS row stride (floats) -> conflict-free A loads

typedef float v2f __attribute__((ext_vector_type(2)));
typedef float v8f __attribute__((ext_vector_type(8)));

// ---------- Phase A: H[j][pos] = sin(w2 @ sin(w1*p + b1) + b2), j-major ----------
__global__ void ck_gen_h(const float* __restrict__ pos_rel,
                         const float* __restrict__ w1, const float* __restrict__ b1,
                         const float* __restrict__ w2, const float* __restrict__ b2,
                         float* __restrict__ H) {
  int pos = blockIdx.x * blockDim.x + threadIdx.x;
  if (pos >= L_) return;
  float p = pos_rel[pos];
  float h1[D_];
  #pragma unroll
  for (int m = 0; m < D_; ++m) h1[m] = sinf(fmaf(w1[m], p, b1[m]));
  for (int j = 0; j < D_; ++j) {
    float s = b2[j];
    #pragma unroll
    for (int m = 0; m < D_; ++m) s = fmaf(w2[j * D_ + m], h1[m], s);
    H[j * L_ + pos] = sinf(s);
  }
}

// ---------- Phase B: Kfl[row][d] = (w3 @ H + b3) at pos = 2048 - d (pre-flipped, padded rows) ----------
__global__ void ck_gen_kfl(const float* __restrict__ H,
                           const float* __restrict__ w3, const float* __restrict__ b3,
                           float* __restrict__ Kfl) {
  int d   = blockIdx.x * blockDim.x + threadIdx.x;
  int row = blockIdx.y;
  if (d >= L_) return;
  int pos = (L_ - 1) - d;
  float s = b3[row];
  #pragma unroll
  for (int j = 0; j < D_; ++j) s = fmaf(w3[row * D_ + j], H[j * L_ + pos], s);
  Kfl[row * LPAD + d] = s;
}

// ---------- Phase C: causal long conv via V_WMMA_F32_16X16X4_F32 ----------
// out[b,o,t] = bias[o] + sum_i sum_{d<=t} Kf[o,i,d] * x[b,i,t-d]
__global__ __launch_bounds__(256)
void ck_conv(const float* __restrict__ x, const float* __restrict__ Kfl,
             const float* __restrict__ bias, float* __restrict__ out) {
  __shared__ float kbuf[CO * KST];   // 32 rows x 64 cols, padded stride: 8704 B
  __shared__ float xrev[TT + DC];    // reversed x window, 128 floats

  const int blk = blockIdx.x;
  const int b   = blk >> 5;            // T_/TT == 32 tiles per batch
  const int jt  = 31 - (blk & 31);     // heavy tiles launch first (work ~ jt+1)
  const int t0  = jt * TT;

  const int tid    = threadIdx.x;
  const int lane   = tid & 31;
  const int wv     = tid >> 5;        // 8 waves
  const int o_base = (wv & 1) << 4;   // 0 / 16
  const int t_sub  = wv >> 1;         // 0..3
  const int n      = lane & 15;       // N index (t within 16-tile)
  const int h      = lane >> 4;       // lane half -> K pair selector

  v8f acc = {};

  const int nchunk = jt + 1;          // causal: only d0 <= t0+63 contributes
  for (int ch = 0; ch < nchunk; ++ch) {
    const int d0   = ch * DC;
    const int xghi = t0 + (TT - 1) - d0;   // xrev[j] = x[b,i, xghi - j]
    for (int i = 0; i < CI; ++i) {
      __syncthreads();
      // stage Kf[0..31, i, d0..d0+63]: 512 float4 loads (16B-aligned via LPAD)
      #pragma unroll
      for (int s = 0; s < 2; ++s) {
        int idx4 = tid + (s << 8);           // 0..511
        int o    = idx4 >> 4;                // 0..31
        int dc4  = (idx4 & 15) << 2;         // 0,4,...,60
        const float4* gp =
            (const float4*)&Kfl[(o * CI + i) * LPAD + d0 + dc4];
        *(float4*)&kbuf[o * KST + dc4] = *gp;
      }
      // stage reversed x window: xrev[j] = x[b,i, xghi - j] (zero outside [0,T))
      if (tid < TT + DC) {
        int g = xghi - tid;
        xrev[tid] = (g >= 0 && g < T_) ? x[(b * CI + i) * T_ + g] : 0.0f;
      }
      __syncthreads();

      // A frag: lane(l) = Kf[o_base + (l&15)][c0..c0+1], c0 = 4k + 2h
      // B frag: lane(l) = {xrev[r], xrev[r+1]}, r = rbase + 4k (increasing -> no swap)
      const int abase = (o_base + n) * KST + (h << 1);
      const int rbase = (TT - 1) - (t_sub << 4) - n + (h << 1);
      #pragma unroll
      for (int k4 = 0; k4 < DC / 4; ++k4) {
        v2f a;
        a.x = kbuf[abase + 4 * k4];
        a.y = kbuf[abase + 4 * k4 + 1];
        const int r = rbase + 4 * k4;
        v2f bm;
        bm.x = xrev[r];
        bm.y = xrev[r + 1];
        acc = __builtin_amdgcn_wmma_f32_16x16x4_f32(
            /*neg_a=*/false, a, /*neg_b=*/false, bm,
            /*c_mod=*/(short)0, acc, /*reuse_a=*/false, /*reuse_b=*/false);
      }
    }
  }

  // C/D layout: VGPR r, lane half h -> M = r + 8h ; N = lane&15
  const int t = t0 + (t_sub << 4) + n;
  #pragma unroll
  for (int r = 0; r < 8; ++r) {
    int o = o_base + r + (h << 3);
    out[((b * CO + o) * T_) + t] = acc[r] + bias[o];
  }
}

extern "C" void kernel_launch(void* const* d_in, const int* in_sizes, int n_in,
                              void* d_out, int out_size, void* d_ws, size_t ws_size,
                              hipStream_t stream) {
  const float* x    = (const float*)d_in[0];
  const float* pos  = (const float*)d_in[1];
  const float* w1   = (const float*)d_in[2];
  const float* b1   = (const float*)d_in[3];
  const float* w2   = (const float*)d_in[4];
  const float* b2   = (const float*)d_in[5];
  const float* w3   = (const float*)d_in[6];
  const float* b3   = (const float*)d_in[7];
  const float* bias = (const float*)d_in[8];
  float* out = (float*)d_out;

  // workspace: Kfl (1024*LPAD floats = 8.4 MB) then H (32*L floats = 0.26 MB)
  float* Kfl = (float*)d_ws;
  float* H   = Kfl + (size_t)(CO * CI) * LPAD;

  ck_gen_h<<<(L_ + 127) / 128, 128, 0, stream>>>(pos, w1, b1, w2, b2, H);
  ck_gen_kfl<<<dim3((L_ + 255) / 256, CO * CI), 256, 0, stream>>>(H, w3, b3, Kfl);
  ck_conv<<<B_ * (T_ / TT), 256, 0, stream>>>(x, Kfl, bias, out);
}